// EmbracementLayer_86354612453674
// MI455X (gfx1250) — compile-verified
//
#include <hip/hip_runtime.h>

// out[b, e] = x[b, idx[e]+1, e]
//   x   : (bs, seq_plus1, emb) float32
//   idx : (emb,)               int32 in [0, seq_plus1-1)
//   out : (bs, emb)            float32
//
// Pure gather: latency-bound. Coalesced idx loads + output stores; scattered
// x reads issued 4-deep per lane with non-temporal hints (no reuse of any
// gathered cacheline, so keep it out of the caches).

__global__ __launch_bounds__(256)
void EmbracementLayer_gather_kernel(const float* __restrict__ x,
                                    const int*   __restrict__ idx,
                                    float*       __restrict__ out,
                                    int emb,
                                    long long batch_stride,   // seq_plus1 * emb
                                    int bs,
                                    int b_per_block) {
    const int e = blockIdx.x * blockDim.x + threadIdx.x;
    if (e >= emb) return;

    // idx[e] is reused across all batches this thread serves -> regular load
    // (hits L2/WGP$ for the other b-blocks). Row r = idx[e] + 1 of the
    // seq dimension; column e within that row.
    const int       r   = idx[e] + 1;
    const long long col = (long long)r * (long long)emb + (long long)e;

    const int b0   = blockIdx.y * b_per_block;
    const int bEnd = (b0 + b_per_block < bs) ? (b0 + b_per_block) : bs;

    // 4 independent scattered loads in flight per lane; NT hint: each
    // gathered cacheline is touched exactly once.
    #pragma unroll 4
    for (int b = b0; b < bEnd; ++b) {
        const float v = __builtin_nontemporal_load(x + (long long)b * batch_stride + col);
        __builtin_nontemporal_store(v, out + (long long)b * (long long)emb + e);
    }
}

extern "C" void kernel_launch(void* const* d_in, const int* in_sizes, int n_in,
                              void* d_out, int out_size, void* d_ws, size_t ws_size,
                              hipStream_t stream) {
    const float* x   = (const float*)d_in[0];  // (bs, seq_plus1, emb) fp32
    const int*   idx = (const int*)  d_in[1];  // (emb,) int32
    float*       out = (float*)d_out;          // (bs, emb) fp32

    const int emb = in_sizes[1];               // 768
    const int bs  = out_size / emb;            // 64
    const long long seq_plus1 = (long long)in_sizes[0] / ((long long)bs * emb); // 4097
    const long long batch_stride = seq_plus1 * (long long)emb;

    const int threads     = 256;
    const int b_per_block = 4;
    dim3 grid((emb + threads - 1) / threads,        // 3
              (bs + b_per_block - 1) / b_per_block, // 16
              1);

    EmbracementLayer_gather_kernel<<<grid, dim3(threads, 1, 1), 0, stream>>>(
        x, idx, out, emb, batch_stride, bs, b_per_block);
}